// Model_60713657696910
// MI455X (gfx1250) — compile-verified
//
#include <hip/hip_runtime.h>
#include <hip/hip_bf16.h>
#include <stdint.h>

// Problem constants (from the reference): N=32768 rows, 2H=4096, H=2048, G=8.
#define TWO_H   4096
#define HALF_H  2048
#define NGROUP  8
#define THREADS 256
#define ROWS_PER_BLOCK 8
#define PER_T   (HALF_H / THREADS)   // 8 outputs per thread per row

#if defined(__has_builtin)
#  if __has_builtin(__builtin_amdgcn_tensor_load_to_lds) && __has_builtin(__builtin_amdgcn_s_wait_tensorcnt)
#    define HAS_TDM 1
#  endif
#endif

typedef unsigned int u32x4 __attribute__((ext_vector_type(4)));
typedef int          i32x4 __attribute__((ext_vector_type(4)));
typedef int          i32x8 __attribute__((ext_vector_type(8)));
typedef float        f32x4 __attribute__((ext_vector_type(4)));  // native vec for NT stores

#ifdef HAS_TDM
// Issue a TDM load of one 4096-int32 row (16 KiB) from global into LDS.
// D# layout per CDNA5 ISA §8 (group0: count/lds_addr/global_addr/type=2,
// group1: data_size=4B, tensor_dim0=tile_dim0=4096, tile_dim1=1, stride0=4096).
__device__ __forceinline__ void tdm_load_row(const int* gptr, unsigned lds_byte_off) {
  unsigned long long ga = (unsigned long long)(uintptr_t)gptr;
  u32x4 g0;
  g0[0] = 1u;                                            // count=1, user mode
  g0[1] = lds_byte_off;                                  // lds_addr (bytes)
  g0[2] = (unsigned)ga;                                  // global_addr[31:0]
  g0[3] = ((unsigned)(ga >> 32) & 0x01FFFFFFu)           // global_addr[56:32]
          | (2u << 30);                                  // type=2 ("image")
  i32x8 g1;
  g1[0] = 2 << 16;                                       // data_size=4B; no mask/pad/iterate
  g1[1] = (TWO_H & 0xFFFF) << 16;                        // tensor_dim0 lo16 @ bits[63:48]
  g1[2] = (TWO_H >> 16) | (1 << 16);                     // tensor_dim0 hi16 | tensor_dim1=1 lo16
  g1[3] = (TWO_H << 16);                                 // tensor_dim1 hi16 | tile_dim0=4096
  g1[4] = 1;                                             // tile_dim1=1, tile_dim2=0
  g1[5] = TWO_H;                                         // tensor_dim0_stride lo32
  g1[6] = 0;                                             // stride0 hi16 | stride1 lo16
  g1[7] = 0;
  i32x4 gz = {0, 0, 0, 0};                               // <=2D tensor: groups 2/3 unused
#if __clang_major__ >= 23
  i32x8 gz8 = {0, 0, 0, 0, 0, 0, 0, 0};
  __builtin_amdgcn_tensor_load_to_lds(g0, g1, gz, gz, gz8, 0);
#else
  __builtin_amdgcn_tensor_load_to_lds(g0, g1, gz, gz, 0);
#endif
}
#endif

__global__ __launch_bounds__(THREADS, 1)
void Model_60713657696910_kernel(const int* __restrict__ x,
                                 const float* __restrict__ wscale,
                                 const float* __restrict__ ascale,
                                 const float* __restrict__ bias,
                                 const float* __restrict__ qscale,
                                 const int* __restrict__ gsize,
                                 float* __restrict__ out, int nrows) {
  __shared__ int   xs[2][TWO_H];          // 32 KiB double buffer
  __shared__ float wred[THREADS / 32];    // per-wave abs-max partials
  __shared__ float s_scale;

  const int tid  = threadIdx.x;
  const int row0 = blockIdx.x * ROWS_PER_BLOCK;
  int nr = nrows - row0;
  if (nr <= 0) return;
  if (nr > ROWS_PER_BLOCK) nr = ROWS_PER_BLOCK;

  // Ragged group boundaries: cum[g] = sum(gsize[0..g])
  int cum[NGROUP];
  {
    int c = 0;
#pragma unroll
    for (int g = 0; g < NGROUP; ++g) { c += gsize[g]; cum[g] = c; }
  }

  const bool issuer = (tid >> 5) == 0;    // wave 0 issues all TDM ops

#ifdef HAS_TDM
  if (issuer) tdm_load_row(x + (size_t)row0 * TWO_H, (unsigned)(uintptr_t)&xs[0][0]);
#endif

  for (int i = 0; i < nr; ++i) {
    const int row = row0 + i;
    const int buf = i & 1;

#ifdef HAS_TDM
    if (issuer) {
      if (i + 1 < nr) {
        tdm_load_row(x + (size_t)(row + 1) * TWO_H,
                     (unsigned)(uintptr_t)&xs[buf ^ 1][0]);
        __builtin_amdgcn_s_wait_tensorcnt(1);   // tile i done, tile i+1 in flight
      } else {
        __builtin_amdgcn_s_wait_tensorcnt(0);   // drain
      }
    }
    __syncthreads();
#else
    {   // fallback: cooperative staged copy
      const int4* src = (const int4*)(x + (size_t)row * TWO_H);
      int4* dst = (int4*)&xs[buf][0];
      for (int j = tid; j < TWO_H / 4; j += THREADS) dst[j] = src[j];
    }
    __syncthreads();
#endif

    // gid = searchsorted(cumsum, row, 'right')
    int gid = 0;
#pragma unroll
    for (int g = 0; g < NGROUP - 1; ++g) gid += (row >= cum[g]) ? 1 : 0;

    const float a = ascale[row];
    const float q = qscale[gid];
    const float* __restrict__ wsr = wscale + (size_t)gid * TWO_H;

    const int cbase = tid * PER_T;
    float vals[PER_T];
    float mx = 0.0f;

#pragma unroll
    for (int v = 0; v < PER_T / 4; ++v) {
      const int cg = cbase + v * 4;
      const int cu = cg + HALF_H;
      int4   xg = *(const int4*)&xs[buf][cg];
      int4   xu = *(const int4*)&xs[buf][cu];
      float4 bg = *(const float4*)(bias + cg);
      float4 bu = *(const float4*)(bias + cu);
      float4 wg = *(const float4*)(wsr + cg);
      float4 wu = *(const float4*)(wsr + cu);
      float gate[4] = { ((float)xg.x + bg.x) * wg.x * a,
                        ((float)xg.y + bg.y) * wg.y * a,
                        ((float)xg.z + bg.z) * wg.z * a,
                        ((float)xg.w + bg.w) * wg.w * a };
      float up[4]   = { ((float)xu.x + bu.x) * wu.x * a,
                        ((float)xu.y + bu.y) * wu.y * a,
                        ((float)xu.z + bu.z) * wu.z * a,
                        ((float)xu.w + bu.w) * wu.w * a };
#pragma unroll
      for (int k = 0; k < 4; ++k) {
        float u = up[k];
        float o = (u / (1.0f + __expf(-u))) * gate[k] * q;  // silu(up)*gate*qs
        vals[v * 4 + k] = o;
        mx = fmaxf(mx, fabsf(o));
      }
    }

    // wave32 abs-max reduce, then cross-wave combine
#pragma unroll
    for (int off = 16; off > 0; off >>= 1)
      mx = fmaxf(mx, __shfl_xor(mx, off, 32));
    if ((tid & 31) == 0) wred[tid >> 5] = mx;
    __syncthreads();
    if (tid == 0) {
      float m = wred[0];
#pragma unroll
      for (int w = 1; w < THREADS / 32; ++w) m = fmaxf(m, wred[w]);
      s_scale = 127.0f / m;
    }
    __syncthreads();
    const float s = s_scale;

    float* __restrict__ orow = out + (size_t)row * HALF_H + cbase;
#pragma unroll
    for (int v = 0; v < PER_T / 4; ++v) {
      f32x4 o;
      o.x = __builtin_rintf(fminf(fmaxf(vals[v * 4 + 0] * s, -128.0f), 127.0f));
      o.y = __builtin_rintf(fminf(fmaxf(vals[v * 4 + 1] * s, -128.0f), 127.0f));
      o.z = __builtin_rintf(fminf(fmaxf(vals[v * 4 + 2] * s, -128.0f), 127.0f));
      o.w = __builtin_rintf(fminf(fmaxf(vals[v * 4 + 3] * s, -128.0f), 127.0f));
      __builtin_nontemporal_store(o, (f32x4*)(orow + v * 4));  // streamed, TH=NT
    }
    __syncthreads();   // all reads of xs[buf] done before it is refilled at i+2
  }
}

extern "C" void kernel_launch(void* const* d_in, const int* in_sizes, int n_in,
                              void* d_out, int out_size, void* d_ws, size_t ws_size,
                              hipStream_t stream) {
  const int*   x      = (const int*)d_in[0];    // x_tensor   [N, 2H] int32
  const float* wscale = (const float*)d_in[1];  // weight_scale [G, 2H] f32
  const float* ascale = (const float*)d_in[2];  // activate_scale [N,1] f32
  const float* bias   = (const float*)d_in[3];  // bias [2H] f32
  const float* qscale = (const float*)d_in[4];  // quant_scale [G] f32
  const int*   gsize  = (const int*)d_in[5];    // group_index [G] int32
  float*       out    = (float*)d_out;          // [N, H] (int8 values as float)

  const int nrows = in_sizes[0] / TWO_H;
  const int grid  = (nrows + ROWS_PER_BLOCK - 1) / ROWS_PER_BLOCK;
  Model_60713657696910_kernel<<<grid, THREADS, 0, stream>>>(
      x, wscale, ascale, bias, qscale, gsize, out, nrows);
}